// AreaAttention_59399397704270
// MI455X (gfx1250) — compile-verified
//
#include <hip/hip_runtime.h>

typedef float v2f   __attribute__((ext_vector_type(2)));
typedef float v8f   __attribute__((ext_vector_type(8)));
typedef float f32x4 __attribute__((ext_vector_type(4)));  // raw vector (builtin-friendly)
typedef int   i32x4_g __attribute__((vector_size(16)));   // matches async builtin's V4i

#define C_    64
#define H_    256
#define W_    256
#define TW    64          // tile columns
#define TP    256         // tile pixels (4 rows * 64 cols)

// LDS layout (float offsets)
#define XS_OFF    0        // x tile       [64][256]
#define VS_OFF    16384    // v tile       [64][256]
#define QKS_OFF   32768    // qk tile      [16][256] (rows 0-7 q, 8-15 k)
#define WVS_OFF   36864    // Wv           [64][64]
#define QKW_OFF   40960    // Wq|Wk packed [16][64]
#define BQK_OFF   41984    // bq|bk        [16]
#define BVV_OFF   42000    // bv           [64]
#define PHS_OFF   42064    // horiz probs  [64 cols][4i][4j]
#define PVS_OFF   43088    // vert  probs  [4 rows][16 cb][4i][4j]
#define SMEM_FLOATS 44112

#if __has_builtin(__builtin_amdgcn_global_load_async_to_lds_b128)
#define HAVE_ASYNC_LDS 1
#else
#define HAVE_ASYNC_LDS 0
#endif

#if HAVE_ASYNC_LDS
// DMA 16B from global straight into LDS (no VGPR round trip); ASYNCcnt-tracked.
#define ASYNC_CP16(dst_lds, src_glb)                                           \
    __builtin_amdgcn_global_load_async_to_lds_b128(                            \
        (__attribute__((address_space(1))) i32x4_g*)(src_glb),                 \
        (__attribute__((address_space(3))) i32x4_g*)(dst_lds), 0, 0)

static __device__ __forceinline__ void wait_async_zero() {
#if __has_builtin(__builtin_amdgcn_s_wait_asynccnt)
    __builtin_amdgcn_s_wait_asynccnt(0);
#else
    asm volatile("s_wait_asynccnt 0x0" ::: "memory");
#endif
}
#endif

__global__ __launch_bounds__(256, 1)
void area_attn_fused(const float* __restrict__ xg,
                     const float* __restrict__ Wq, const float* __restrict__ bq,
                     const float* __restrict__ Wk, const float* __restrict__ bk,
                     const float* __restrict__ Wv, const float* __restrict__ bv,
                     const float* __restrict__ gamma,
                     float* __restrict__ outg)
{
    extern __shared__ float smem[];
    float* xs  = smem + XS_OFF;
    float* vsm = smem + VS_OFF;
    float* qks = smem + QKS_OFF;
    float* wvs = smem + WVS_OFF;
    float* qkw = smem + QKW_OFF;
    float* bqk = smem + BQK_OFF;
    float* bvv = smem + BVV_OFF;
    float* phs = smem + PHS_OFF;
    float* pvs = smem + PVS_OFF;

    const int tid = threadIdx.x;                 // 0..255
    const int blk = blockIdx.x;                  // 0..2047
    const int b   = blk >> 8;                    // batch
    const int rem = blk & 255;
    const int h0  = (rem >> 2) * 4;              // row-block origin
    const int w0  = (rem & 3) * 64;              // col-span origin

    // ---------------- Phase 0: stage x tile + weights into LDS ----------------
    {
        const float* xbase = xg + (((size_t)b * C_) * H_ + h0) * W_ + w0;
#if HAVE_ASYNC_LDS
        // x tile: 4096 x 16B transfers, DMA'd directly into LDS.
        #pragma unroll
        for (int it = 0; it < 16; ++it) {
            int q4  = it * 256 + tid;            // float4 id 0..4095
            int seg = q4 >> 4;                   // 0..255 == c*4 + r
            int c   = seg >> 2;
            int r   = seg & 3;
            int col = (q4 & 15) << 2;
            ASYNC_CP16(&xs[c * TP + r * TW + col],
                       xbase + (size_t)c * H_ * W_ + (size_t)r * W_ + col);
        }
        // Wv: 1024 x 16B
        #pragma unroll
        for (int it = 0; it < 4; ++it) {
            int q4 = it * 256 + tid;
            ASYNC_CP16(&wvs[q4 << 2], Wv + ((size_t)q4 << 2));
        }
        // Wq|Wk packed: 256 x 16B
        {
            int row = tid >> 4;                  // 0..15
            int col = (tid & 15) << 2;
            const float* src = (row < 8) ? (Wq + row * 64 + col)
                                         : (Wk + (row - 8) * 64 + col);
            ASYNC_CP16(&qkw[row * 64 + col], src);
        }
#else
        #pragma unroll
        for (int it = 0; it < 16; ++it) {
            int q4  = it * 256 + tid;
            int seg = q4 >> 4;
            int c   = seg >> 2;
            int r   = seg & 3;
            int col = (q4 & 15) << 2;
            f32x4 val = __builtin_nontemporal_load(reinterpret_cast<const f32x4*>(
                xbase + (size_t)c * H_ * W_ + (size_t)r * W_ + col));
            *reinterpret_cast<f32x4*>(&xs[c * TP + r * TW + col]) = val;
        }
        #pragma unroll
        for (int it = 0; it < 4; ++it) {
            int q4 = it * 256 + tid;
            *reinterpret_cast<f32x4*>(&wvs[q4 << 2]) =
                *reinterpret_cast<const f32x4*>(Wv + ((size_t)q4 << 2));
        }
        {
            int row = tid >> 4;
            int col = (tid & 15) << 2;
            const float* src = (row < 8) ? (Wq + row * 64 + col)
                                         : (Wk + (row - 8) * 64 + col);
            *reinterpret_cast<f32x4*>(&qkw[row * 64 + col]) =
                *reinterpret_cast<const f32x4*>(src);
        }
#endif
        if (tid < 16) bqk[tid] = (tid < 8) ? bq[tid] : bk[tid - 8];
        if (tid < 64) bvv[tid] = bv[tid];
#if HAVE_ASYNC_LDS
        wait_async_zero();
#endif
    }
    __syncthreads();

    // ------------- Phase 1: q/k/v projections via V_WMMA_F32_16X16X4_F32 -------------
    {
        const int lane  = tid & 31;
        const int wave  = tid >> 5;
        const int l16   = lane & 15;
        const int lhalf = lane >> 4;             // 0 | 1

        #pragma unroll
        for (int t = 0; t < 2; ++t) {
            const int nt  = wave * 2 + t;        // N-tile 0..15
            const int pix = nt * 16 + l16;       // this lane's pixel (B/C column)
            v8f aqk = {};
            v8f av0 = {}, av1 = {}, av2 = {}, av3 = {};
            #pragma unroll
            for (int kk = 0; kk < 16; ++kk) {
                const int kA = kk * 4 + lhalf * 2;   // lane-half K offset (A/B layout)
                v2f bf;                               // B: X[k][pix], X[k+1][pix]
                bf.x = xs[kA * TP + pix];
                bf.y = xs[(kA + 1) * TP + pix];
                v2f af;                               // A: W[m][k], W[m][k+1]
                af.x = qkw[l16 * 64 + kA];
                af.y = qkw[l16 * 64 + kA + 1];
                aqk = __builtin_amdgcn_wmma_f32_16x16x4_f32(false, af, false, bf, (short)0, aqk, false, false);
                af.x = wvs[l16 * 64 + kA];
                af.y = wvs[l16 * 64 + kA + 1];
                av0 = __builtin_amdgcn_wmma_f32_16x16x4_f32(false, af, false, bf, (short)0, av0, false, false);
                af.x = wvs[(16 + l16) * 64 + kA];
                af.y = wvs[(16 + l16) * 64 + kA + 1];
                av1 = __builtin_amdgcn_wmma_f32_16x16x4_f32(false, af, false, bf, (short)0, av1, false, false);
                af.x = wvs[(32 + l16) * 64 + kA];
                af.y = wvs[(32 + l16) * 64 + kA + 1];
                av2 = __builtin_amdgcn_wmma_f32_16x16x4_f32(false, af, false, bf, (short)0, av2, false, false);
                af.x = wvs[(48 + l16) * 64 + kA];
                af.y = wvs[(48 + l16) * 64 + kA + 1];
                av3 = __builtin_amdgcn_wmma_f32_16x16x4_f32(false, af, false, bf, (short)0, av3, false, false);
            }
            // D layout: VGPR r -> M=r (lanes 0-15) or M=r+8 (lanes 16-31), N=l16
            const int mb = lhalf * 8;
            #pragma unroll
            for (int r = 0; r < 8; ++r) {
                const int m = mb + r;
                qks[m * TP + pix]          = aqk[r] + bqk[m];
                vsm[m * TP + pix]          = av0[r] + bvv[m];
                vsm[(16 + m) * TP + pix]   = av1[r] + bvv[16 + m];
                vsm[(32 + m) * TP + pix]   = av2[r] + bvv[32 + m];
                vsm[(48 + m) * TP + pix]   = av3[r] + bvv[48 + m];
            }
        }
    }
    __syncthreads();

    // ------------- Phase 2: 4x4 attention scores + softmax (tiny, VALU) -------------
    if (tid < 128) {
        int pbase[4];
        float* Pdst;
        if (tid < 64) {                          // horizontal: one column each
            const int w = tid;
            #pragma unroll
            for (int i = 0; i < 4; ++i) pbase[i] = i * TW + w;
            Pdst = &phs[w * 16];
        } else {                                 // vertical: one (row, col-block) each
            const int idx  = tid - 64;           // 0..63
            const int r    = idx >> 4;
            const int cb   = idx & 15;
            const int base = r * TW + cb * 4;
            #pragma unroll
            for (int i = 0; i < 4; ++i) pbase[i] = base + i;
            Pdst = &pvs[idx * 16];
        }
        float S[4][4];
        #pragma unroll
        for (int i = 0; i < 4; ++i)
            #pragma unroll
            for (int j = 0; j < 4; ++j) {
                float s = 0.f;
                #pragma unroll
                for (int c = 0; c < 8; ++c)
                    s += qks[c * TP + pbase[i]] * qks[(c + 8) * TP + pbase[j]];
                S[i][j] = s;
            }
        #pragma unroll
        for (int i = 0; i < 4; ++i) {            // softmax over key dim j
            float m  = fmaxf(fmaxf(S[i][0], S[i][1]), fmaxf(S[i][2], S[i][3]));
            float e0 = __expf(S[i][0] - m), e1 = __expf(S[i][1] - m);
            float e2 = __expf(S[i][2] - m), e3 = __expf(S[i][3] - m);
            float inv = 1.f / (e0 + e1 + e2 + e3);
            Pdst[i * 4 + 0] = e0 * inv;
            Pdst[i * 4 + 1] = e1 * inv;
            Pdst[i * 4 + 2] = e2 * inv;
            Pdst[i * 4 + 3] = e3 * inv;
        }
    }
    __syncthreads();

    // ------------- Phase 3: fused output  out = gamma*(h+v)+x -------------
    {
        const float g  = gamma[0];
        const int p    = tid;                    // one pixel per thread
        const int w    = p & 63;
        const int rr   = p >> 6;                 // row-in-block == output j (horiz)
        const int cb   = w >> 2;
        const int u    = w & 3;                  // col-in-block == output j (vert)
        const float* PH = &phs[w * 16];
        const float* PV = &pvs[(rr * 16 + cb) * 16];
        const float ph0 = PH[0 + rr], ph1 = PH[4 + rr], ph2 = PH[8 + rr], ph3 = PH[12 + rr];
        const float pv0 = PV[0 + u],  pv1 = PV[4 + u],  pv2 = PV[8 + u],  pv3 = PV[12 + u];
        const int vb = rr * TW + cb * 4;
        float* op = outg + (((size_t)b * C_) * H_ + (h0 + rr)) * W_ + w0 + w;
        #pragma unroll 4
        for (int c = 0; c < 64; ++c) {
            const float* vr = &vsm[c * TP];
            float hsum = vr[w]            * ph0 + vr[TW + w]      * ph1
                       + vr[2 * TW + w]   * ph2 + vr[3 * TW + w]  * ph3;
            float vsum = vr[vb] * pv0 + vr[vb + 1] * pv1
                       + vr[vb + 2] * pv2 + vr[vb + 3] * pv3;
            // streamed once, never re-read: bypass-temporal store
            __builtin_nontemporal_store(g * (hsum + vsum) + xs[c * TP + p],
                                        op + (size_t)c * H_ * W_);
        }
    }
}

extern "C" void kernel_launch(void* const* d_in, const int* in_sizes, int n_in,
                              void* d_out, int out_size, void* d_ws, size_t ws_size,
                              hipStream_t stream) {
    (void)in_sizes; (void)n_in; (void)d_ws; (void)ws_size; (void)out_size;
    const float* x  = (const float*)d_in[0];
    const float* Wq = (const float*)d_in[1];
    const float* bq = (const float*)d_in[2];
    const float* Wk = (const float*)d_in[3];
    const float* bk = (const float*)d_in[4];
    const float* Wv = (const float*)d_in[5];
    const float* bv = (const float*)d_in[6];
    const float* gm = (const float*)d_in[7];
    float* out = (float*)d_out;

    dim3 grid(2048);   // 8 batches * 64 row-blocks * 4 col-spans
    dim3 block(256);   // 8 wave32
    size_t smem = (size_t)SMEM_FLOATS * sizeof(float);
    hipLaunchKernelGGL(area_attn_fused, grid, block, smem, stream,
                       x, Wq, bq, Wk, bk, Wv, bv, gm, out);
}